// RTWAgent_64914135712080
// MI455X (gfx1250) — compile-verified
//
#include <hip/hip_runtime.h>
#include <math.h>

typedef __attribute__((ext_vector_type(2))) float f32x2;
typedef __attribute__((ext_vector_type(8))) float f32x8;

#define A_N   8
#define NACT_ 16
#define R_DIM 64
#define OBS_D 128
#define HID_D 256
#define INP_D 152
#define EPN   2048
#define ROWS  (EPN * A_N)     /* 16384 */
#define TROWS (ROWS * A_N)    /* 131072 */

/* d_out layout (floats): q_reflect | h | loss1 | loss2 */
#define H_OFF  (ROWS * NACT_)            /* 262144  */
#define L1_OFF (H_OFF + ROWS * R_DIM)    /* 1310720 */
#define L2_OFF (L1_OFF + 1)

/* -------- fp32 WMMA 16x16x4 (wave32) --------
   A 16x4: lane m = lane&15; VGPR v holds K = v + 2*(lane>>4)
   B 4x16: lane n = lane&15; same K mapping
   C/D 16x16: VGPR q holds (M = q + 8*(lane>>4), N = lane&15)          */
__device__ __forceinline__ f32x8 wmma4(f32x2 a, f32x2 b, f32x8 c) {
  return __builtin_amdgcn_wmma_f32_16x16x4_f32(false, a, false, b, (short)0, c,
                                               false, false);
}

/* single-accumulator GEMM K-loop (A + one B row-tile) */
template <typename FA>
__device__ __forceinline__ f32x8 gemm_k(int K, int k2, FA loadA,
                                        const float* __restrict__ brow, f32x8 c) {
  for (int kb = 0; kb < K; kb += 4) {
    int k0 = kb + k2;
    f32x2 a, b;
    a.x = loadA(k0);
    a.y = loadA(k0 + 1);
    b.x = brow[k0];
    b.y = brow[k0 + 1];
    c = wmma4(a, b, c);
  }
  return c;
}

/* multi-accumulator GEMM: one K pass, A fragment loaded once per K-step and
   reused across NT output tiles; B offsets fold to immediate offsets. */
template <int NT, int K, typename FA>
__device__ __forceinline__ void gemm_multi(int k2, FA loadA,
                                           const float* __restrict__ bbase,
                                           f32x8* acc) {
  for (int kb = 0; kb < K; kb += 4) {
    const int k0 = kb + k2;
    f32x2 a;
    a.x = loadA(k0);
    a.y = loadA(k0 + 1);
#pragma unroll
    for (int nt = 0; nt < NT; ++nt) {
      f32x2 b;
      b.x = bbase[nt * 16 * K + k0];
      b.y = bbase[nt * 16 * K + k0 + 1];
      acc[nt] = wmma4(a, b, acc[nt]);
    }
  }
}

__device__ __forceinline__ float waveRedSum(float v) {
#pragma unroll
  for (int s = 16; s > 0; s >>= 1) v += __shfl_xor(v, s, 32);
  return v;
}

/* ---------------- init: zero the two scalar losses ---------------- */
__global__ void k_init(float* out) {
  out[L1_OFF] = 0.f;
  out[L2_OFF] = 0.f;
}

/* ---------------- fused fc1 + GRU cell ---------------- */
__global__ __launch_bounds__(128) void k_gru(
    const float* __restrict__ inp, const float* __restrict__ hin,
    const float* __restrict__ fc1w, const float* __restrict__ fc1b,
    const float* __restrict__ wih, const float* __restrict__ bih,
    const float* __restrict__ whh, const float* __restrict__ bhh,
    float* __restrict__ out) {
  __shared__ float xs[4][16 * 64];
  const int lane = threadIdx.x & 31;
  const int wv = threadIdx.x >> 5;
  const int hf = lane >> 4, l16 = lane & 15;
  const int k2 = hf << 1;
  const int tile = blockIdx.x * 4 + wv;
  const int r0 = tile * 16;
  float* X = xs[wv];

  /* stage 1: x = relu(inputs @ fc1.T + b)  (16 x 64 tile into LDS) */
  {
    const float* arow = inp + (size_t)(r0 + l16) * INP_D;
    f32x8 acc[4] = {};
    gemm_multi<4, INP_D>(k2, [&](int k) { return arow[k]; },
                         fc1w + (size_t)l16 * INP_D, acc);
#pragma unroll
    for (int nt = 0; nt < 4; ++nt) {
      float bb = fc1b[nt * 16 + l16];
#pragma unroll
      for (int q = 0; q < 8; ++q)
        X[(q + 8 * hf) * 64 + nt * 16 + l16] = fmaxf(acc[nt][q] + bb, 0.f);
    }
  }
  __syncthreads();

  /* stage 2: gi = x@Wih.T, gh = h@Whh.T, gate math, h out */
  for (int j = 0; j < 4; ++j) {
    const int n = j * 16 + l16;
    const float* ax = X + l16 * 64;
    const float* ah = hin + (size_t)(r0 + l16) * R_DIM;
    const float* wir = wih + (size_t)(n) * R_DIM;
    const float* wiz = wih + (size_t)(64 + n) * R_DIM;
    const float* win = wih + (size_t)(128 + n) * R_DIM;
    const float* whr = whh + (size_t)(n) * R_DIM;
    const float* whz = whh + (size_t)(64 + n) * R_DIM;
    const float* whn = whh + (size_t)(128 + n) * R_DIM;
    f32x8 cir = {}, ciz = {}, cin = {}, chr_ = {}, chz = {}, chn = {};
    for (int kb = 0; kb < R_DIM; kb += 4) {
      int k0 = kb + k2;
      f32x2 a, b;
      a.x = ax[k0]; a.y = ax[k0 + 1];
      b.x = wir[k0]; b.y = wir[k0 + 1]; cir = wmma4(a, b, cir);
      b.x = wiz[k0]; b.y = wiz[k0 + 1]; ciz = wmma4(a, b, ciz);
      b.x = win[k0]; b.y = win[k0 + 1]; cin = wmma4(a, b, cin);
      a.x = ah[k0]; a.y = ah[k0 + 1];
      b.x = whr[k0]; b.y = whr[k0 + 1]; chr_ = wmma4(a, b, chr_);
      b.x = whz[k0]; b.y = whz[k0 + 1]; chz = wmma4(a, b, chz);
      b.x = whn[k0]; b.y = whn[k0 + 1]; chn = wmma4(a, b, chn);
    }
    const float bir = bih[n], biz = bih[64 + n], bin_ = bih[128 + n];
    const float bhr = bhh[n], bhz = bhh[64 + n], bhn = bhh[128 + n];
#pragma unroll
    for (int q = 0; q < 8; ++q) {
      const int rr = r0 + q + 8 * hf;
      float gr = (cir[q] + bir) + (chr_[q] + bhr);
      float gz = (ciz[q] + biz) + (chz[q] + bhz);
      float rg = 1.f / (1.f + expf(-gr));
      float zg = 1.f / (1.f + expf(-gz));
      float ng = tanhf((cin[q] + bin_) + rg * (chn[q] + bhn));
      float hprev = hin[(size_t)rr * R_DIM + n];
      out[H_OFF + (size_t)rr * R_DIM + n] = (1.f - zg) * ng + zg * hprev;
    }
  }
}

/* ---------------- teammate model + CE loss + masked actions ---------------- */
__device__ __forceinline__ float tmA(const float* __restrict__ hrow, bool zero,
                                     int jA, int k) {
  if (zero) return 0.f;
  if (k < 64) return hrow[k];
  return (k - 64 == jA) ? 1.f : 0.f;
}

__global__ __launch_bounds__(128) void k_teammate(
    const float* __restrict__ hmat, const int* __restrict__ u,
    const float* __restrict__ w1, const float* __restrict__ b1,
    const float* __restrict__ w2, const float* __restrict__ b2,
    float* __restrict__ am, float* __restrict__ lossOut) {
  __shared__ float hs_[4][16 * 256];
  const int lane = threadIdx.x & 31;
  const int wv = threadIdx.x >> 5;
  const int hf = lane >> 4, l16 = lane & 15;
  const int k2 = hf << 1;
  const int tile = blockIdx.x * 4 + wv;
  const int g0 = tile * 16;
  float* HS = hs_[wv];

  const int gm = g0 + l16;
  const int jA = gm & 7, iA = (gm >> 3) & 7, eA = gm >> 6;
  const float* hrow = hmat + (size_t)(eA * 8 + jA) * R_DIM;
  const bool zero = (iA == jA);

  /* layer 1: (R+A=72) -> 256, relu, into LDS (16 N-tiles, single K pass) */
  {
    f32x8 acc[16] = {};
    gemm_multi<16, 72>(k2, [&](int k) { return tmA(hrow, zero, jA, k); },
                       w1 + (size_t)l16 * 72, acc);
#pragma unroll
    for (int nt = 0; nt < 16; ++nt) {
      float bb = b1[nt * 16 + l16];
#pragma unroll
      for (int q = 0; q < 8; ++q)
        HS[(q + 8 * hf) * 256 + nt * 16 + l16] = fmaxf(acc[nt][q] + bb, 0.f);
    }
  }
  __syncthreads();

  /* layer 2: 256 -> 16 */
  f32x8 c = {};
  {
    const float* ar = HS + l16 * 256;
    const float* br = w2 + (size_t)l16 * 256;
    c = gemm_k(256, k2, [&](int k) { return ar[k]; }, br, c);
  }
  const float bb = b2[l16];

  float ceAcc = 0.f;
#pragma unroll
  for (int q = 0; q < 8; ++q) {
    const int g = g0 + q + 8 * hf;
    const int j2 = g & 7, i2 = (g >> 3) & 7, e2 = g >> 6;
    const float logit = c[q] + bb; /* N = l16 */
    const float msk = (i2 != j2) ? 1.f : 0.f;
    am[(size_t)g * NACT_ + l16] = logit * msk;
    /* row-wise log-softmax CE across the 16 lanes of this half */
    float mx = logit;
#pragma unroll
    for (int s = 1; s < 16; s <<= 1) mx = fmaxf(mx, __shfl_xor(mx, s, 32));
    float se = expf(logit - mx);
#pragma unroll
    for (int s = 1; s < 16; s <<= 1) se += __shfl_xor(se, s, 32);
    const int lbl = u[e2 * 8 + j2];
    float sel = (l16 == lbl) ? logit : 0.f;
#pragma unroll
    for (int s = 1; s < 16; s <<= 1) sel += __shfl_xor(sel, s, 32);
    const float ce = (mx + logf(se)) - sel;
    if (l16 == 0) ceAcc += ce * msk;
  }
  ceAcc = waveRedSum(ceAcc);
  if (lane == 0) atomicAdd(lossOut, ceAcc * (1.0f / (float)TROWS));
}

/* ---------------- world model + MSE loss ---------------- */
__global__ __launch_bounds__(128) void k_world(
    const float* __restrict__ obs, const float* __restrict__ obs_next,
    const float* __restrict__ am, const float* __restrict__ w1,
    const float* __restrict__ b1, const float* __restrict__ w2,
    const float* __restrict__ b2, float* __restrict__ ohat,
    float* __restrict__ lossOut) {
  __shared__ float hs_[4][16 * 256];
  const int lane = threadIdx.x & 31;
  const int wv = threadIdx.x >> 5;
  const int hf = lane >> 4, l16 = lane & 15;
  const int k2 = hf << 1;
  const int tile = blockIdx.x * 4 + wv;
  const int r0 = tile * 16;
  float* HS = hs_[wv];

  const float* orow = obs + (size_t)(r0 + l16) * OBS_D;
  const float* arow = am + (size_t)(r0 + l16) * (A_N * NACT_);

  /* layer 1: 256 -> 256, relu (16 N-tiles, single K pass) */
  {
    f32x8 acc[16] = {};
    gemm_multi<16, 256>(
        k2, [&](int k) { return (k < OBS_D) ? orow[k] : arow[k - OBS_D]; },
        w1 + (size_t)l16 * 256, acc);
#pragma unroll
    for (int nt = 0; nt < 16; ++nt) {
      float bb = b1[nt * 16 + l16];
#pragma unroll
      for (int q = 0; q < 8; ++q)
        HS[(q + 8 * hf) * 256 + nt * 16 + l16] = fmaxf(acc[nt][q] + bb, 0.f);
    }
  }
  __syncthreads();

  /* layer 2: 256 -> 128 (8 N-tiles, single K pass), MSE vs obs_next */
  float mseAcc = 0.f;
  {
    const float* ar = HS + l16 * 256;
    f32x8 acc[8] = {};
    gemm_multi<8, 256>(k2, [&](int k) { return ar[k]; },
                       w2 + (size_t)l16 * 256, acc);
#pragma unroll
    for (int nt = 0; nt < 8; ++nt) {
      const int col = nt * 16 + l16;
      const float bb = b2[col];
#pragma unroll
      for (int q = 0; q < 8; ++q) {
        const int rr = r0 + q + 8 * hf;
        float o = acc[nt][q] + bb;
        ohat[(size_t)rr * OBS_D + col] = o;
        float d = o - obs_next[(size_t)rr * OBS_D + col];
        mseAcc += d * d;
      }
    }
  }
  mseAcc = waveRedSum(mseAcc);
  if (lane == 0) atomicAdd(lossOut, mseAcc * (1.0f / (float)(ROWS * OBS_D)));
}

/* ---------------- value network: [h_j | am] 80 -> 64 -> 16 ---------------- */
__global__ __launch_bounds__(128) void k_value(
    const float* __restrict__ hmat, const float* __restrict__ am,
    const float* __restrict__ w1, const float* __restrict__ b1,
    const float* __restrict__ w2, const float* __restrict__ b2,
    float* __restrict__ val) {
  __shared__ float vs_[4][16 * 64];
  const int lane = threadIdx.x & 31;
  const int wv = threadIdx.x >> 5;
  const int hf = lane >> 4, l16 = lane & 15;
  const int k2 = hf << 1;
  const int tile = blockIdx.x * 4 + wv;
  const int g0 = tile * 16;
  float* VS = vs_[wv];

  const int gm = g0 + l16;
  const int jA = gm & 7, eA = gm >> 6;
  const float* hrow = hmat + (size_t)(eA * 8 + jA) * R_DIM; /* hrep: no mask */
  const float* arow = am + (size_t)gm * NACT_;

  {
    f32x8 acc[4] = {};
    gemm_multi<4, 80>(k2,
                      [&](int k) { return (k < 64) ? hrow[k] : arow[k - 64]; },
                      w1 + (size_t)l16 * 80, acc);
#pragma unroll
    for (int nt = 0; nt < 4; ++nt) {
      float bb = b1[nt * 16 + l16];
#pragma unroll
      for (int q = 0; q < 8; ++q)
        VS[(q + 8 * hf) * 64 + nt * 16 + l16] = fmaxf(acc[nt][q] + bb, 0.f);
    }
  }
  __syncthreads();

  f32x8 c = {};
  {
    const float* ar = VS + l16 * 64;
    const float* br = w2 + (size_t)l16 * 64;
    c = gemm_k(64, k2, [&](int k) { return ar[k]; }, br, c);
  }
  const float bb = b2[l16];
#pragma unroll
  for (int q = 0; q < 8; ++q) {
    const int g = g0 + q + 8 * hf;
    val[(size_t)g * NACT_ + l16] = c[q] + bb;
  }
}

/* ---------------- query: [obs | ohat] 256 -> 64 ---------------- */
__global__ __launch_bounds__(128) void k_query(
    const float* __restrict__ obs, const float* __restrict__ ohat,
    const float* __restrict__ wq, const float* __restrict__ bq,
    float* __restrict__ query) {
  const int lane = threadIdx.x & 31;
  const int wv = threadIdx.x >> 5;
  const int hf = lane >> 4, l16 = lane & 15;
  const int k2 = hf << 1;
  const int tile = blockIdx.x * 4 + wv;
  const int r0 = tile * 16;

  const float* orow = obs + (size_t)(r0 + l16) * OBS_D;
  const float* prow = ohat + (size_t)(r0 + l16) * OBS_D;

  f32x8 acc[4] = {};
  gemm_multi<4, 256>(
      k2, [&](int k) { return (k < OBS_D) ? orow[k] : prow[k - OBS_D]; },
      wq + (size_t)l16 * 256, acc);
#pragma unroll
  for (int nt = 0; nt < 4; ++nt) {
    const int col = nt * 16 + l16;
    const float bb = bq[col];
#pragma unroll
    for (int q = 0; q < 8; ++q) {
      const int rr = r0 + q + 8 * hf;
      query[(size_t)rr * 64 + col] = acc[nt][q] + bb;
    }
  }
}

/* ---------------- reflection: keys, masked softmax, weighted value sum ------ */
__global__ __launch_bounds__(256) void k_reflect(
    const float* __restrict__ query, const float* __restrict__ am,
    const float* __restrict__ val, const float* __restrict__ wk,
    const float* __restrict__ bk, float* __restrict__ out) {
  const int r = blockIdx.x * blockDim.x + threadIdx.x;
  if (r >= ROWS) return;
  const int iRow = r & 7;
  const float* qr = query + (size_t)r * 64;
  const float* ar = am + (size_t)r * (A_N * NACT_);

  float s[8];
#pragma unroll
  for (int j = 0; j < 8; ++j) {
    float av[16];
#pragma unroll
    for (int k = 0; k < 16; ++k) av[k] = ar[j * 16 + k];
    float sj = 0.f;
    for (int d = 0; d < 64; ++d) {
      float kd = bk[d];
      const float* wkd = wk + d * 16;
#pragma unroll
      for (int k = 0; k < 16; ++k) kd += wkd[k] * av[k];
      sj += qr[d] * kd;
    }
    s[j] = sj * 0.125f; /* 1/sqrt(64) */
  }
  /* self-exclusion without dynamic register-array indexing */
#pragma unroll
  for (int j = 0; j < 8; ++j)
    if (j == iRow) s[j] = -1e9f;
  float mx = s[0];
#pragma unroll
  for (int j = 1; j < 8; ++j) mx = fmaxf(mx, s[j]);
  float w[8];
  float se = 0.f;
#pragma unroll
  for (int j = 0; j < 8; ++j) {
    w[j] = expf(s[j] - mx);
    se += w[j];
  }
  const float inv = 1.f / se;
#pragma unroll
  for (int n = 0; n < 16; ++n) {
    float acc = 0.f;
#pragma unroll
    for (int j = 0; j < 8; ++j)
      acc += w[j] * inv * val[((size_t)r * 8 + j) * NACT_ + n];
    out[(size_t)r * NACT_ + n] = acc;
  }
}

/* ---------------- host launch ---------------- */
extern "C" void kernel_launch(void* const* d_in, const int* in_sizes, int n_in,
                              void* d_out, int out_size, void* d_ws,
                              size_t ws_size, hipStream_t stream) {
  (void)in_sizes; (void)n_in; (void)out_size; (void)ws_size;
  const float* inputs = (const float*)d_in[0];
  const float* hidden = (const float*)d_in[1];
  const float* obs = (const float*)d_in[2];
  const float* obs_next = (const float*)d_in[3];
  const int* u = (const int*)d_in[4];
  const float* fc1w = (const float*)d_in[5];
  const float* fc1b = (const float*)d_in[6];
  const float* wih = (const float*)d_in[7];
  const float* bih = (const float*)d_in[8];
  const float* whh = (const float*)d_in[9];
  const float* bhh = (const float*)d_in[10];
  const float* tmw1 = (const float*)d_in[11];
  const float* tmb1 = (const float*)d_in[12];
  const float* tmw2 = (const float*)d_in[13];
  const float* tmb2 = (const float*)d_in[14];
  const float* wmw1 = (const float*)d_in[15];
  const float* wmb1 = (const float*)d_in[16];
  const float* wmw2 = (const float*)d_in[17];
  const float* wmb2 = (const float*)d_in[18];
  const float* wqw = (const float*)d_in[19];
  const float* wqb = (const float*)d_in[20];
  const float* wkw = (const float*)d_in[21];
  const float* wkb = (const float*)d_in[22];
  const float* wv1w = (const float*)d_in[23];
  const float* wv1b = (const float*)d_in[24];
  const float* wv2w = (const float*)d_in[25];
  const float* wv2b = (const float*)d_in[26];

  float* out = (float*)d_out;
  float* ws = (float*)d_ws;
  float* am = ws;                                 /* TROWS*16 = 2,097,152 f */
  float* ohat = am + (size_t)TROWS * NACT_;       /* ROWS*128 = 2,097,152 f */
  float* val = ohat + (size_t)ROWS * OBS_D;       /* TROWS*16 = 2,097,152 f */
  float* query = val + (size_t)TROWS * NACT_;     /* ROWS*64  = 1,048,576 f */

  k_init<<<1, 1, 0, stream>>>(out);
  k_gru<<<ROWS / 16 / 4, 128, 0, stream>>>(inputs, hidden, fc1w, fc1b, wih,
                                           bih, whh, bhh, out);
  k_teammate<<<TROWS / 16 / 4, 128, 0, stream>>>(out + H_OFF, u, tmw1, tmb1,
                                                 tmw2, tmb2, am, out + L1_OFF);
  k_world<<<ROWS / 16 / 4, 128, 0, stream>>>(obs, obs_next, am, wmw1, wmb1,
                                             wmw2, wmb2, ohat, out + L2_OFF);
  k_value<<<TROWS / 16 / 4, 128, 0, stream>>>(out + H_OFF, am, wv1w, wv1b,
                                              wv2w, wv2b, val);
  k_query<<<ROWS / 16 / 4, 128, 0, stream>>>(obs, ohat, wqw, wqb, query);
  k_reflect<<<ROWS / 256, 256, 0, stream>>>(query, am, val, wkw, wkb, out);
}